// TwoTower_89507118449226
// MI455X (gfx1250) — compile-verified
//
#include <hip/hip_runtime.h>
#include <hip/hip_bf16.h>

typedef float v2f __attribute__((ext_vector_type(2)));
typedef float v4f __attribute__((ext_vector_type(4)));
typedef float v8f __attribute__((ext_vector_type(8)));

#define UID_DIM 64
#define MID_DIM 64
#define CAT_DIM 16
#define USER_L 32
#define M1DIM 64
#define M2DIM 32
#define MIN_DIM 80        // movie_in = concat(movie 64, cat 16)
#define MAX_CATS 8
#define WAVES 4
#define TPB (WAVES * 32)
#define SSTR 80           // per-wave LDS tile stride (floats)

__device__ __forceinline__ v8f wmma4(v2f a, v2f b, v8f c) {
  // D = A(16x4 f32) * B(4x16 f32) + C(16x16 f32), full fp32 matrix pipe
  return __builtin_amdgcn_wmma_f32_16x16x4_f32(
      /*neg_a=*/false, a, /*neg_b=*/false, b,
      /*c_mod=*/(short)0, c, /*reuse_a=*/false, /*reuse_b=*/false);
}

// B-fragment from K-pair-interleaved weights: sw[(k>>1)*2N + 2*col] = {W[k][col], W[k+1][col]}
__device__ __forceinline__ v2f ldB(const float* sw, int N2, int kk, int col) {
  return *(const v2f*)(sw + (kk >> 1) * N2 + 2 * col);
}

__global__ __launch_bounds__(TPB, 1)
void twotower_kernel(const float* __restrict__ user_emb,
                     const float* __restrict__ movie_emb,
                     const float* __restrict__ cat_emb,
                     const float* __restrict__ Wu, const float* __restrict__ bu,
                     const float* __restrict__ W1, const float* __restrict__ b1,
                     const float* __restrict__ W2, const float* __restrict__ b2,
                     const int* __restrict__ user_idx,
                     const int* __restrict__ movie_idx,
                     const int* __restrict__ cat_lst,
                     float* __restrict__ out, int B)
{
  // weights stored K-pair interleaved: dst = (k>>1)*2N + 2n + (k&1)
  __shared__ float sWu[UID_DIM * USER_L];   // 64x32 swizzled
  __shared__ float sW1[MIN_DIM * M1DIM];    // 80x64 swizzled
  __shared__ float sW2[M1DIM * M2DIM];      // 64x32 swizzled
  __shared__ float sbu[USER_L];
  __shared__ float sb1[M1DIM];
  __shared__ float sb2[M2DIM];
  __shared__ float sbuf[WAVES][16 * SSTR];  // per-wave 16x80 staging tile

  const int t = threadIdx.x;
  for (int i = t; i < UID_DIM * USER_L; i += TPB) {
    int k = i >> 5, n = i & 31;
    sWu[(k >> 1) * 64 + 2 * n + (k & 1)] = Wu[i];
  }
  for (int i = t; i < MIN_DIM * M1DIM; i += TPB) {
    int k = i >> 6, n = i & 63;
    sW1[(k >> 1) * 128 + 2 * n + (k & 1)] = W1[i];
  }
  for (int i = t; i < M1DIM * M2DIM; i += TPB) {
    int k = i >> 5, n = i & 31;
    sW2[(k >> 1) * 64 + 2 * n + (k & 1)] = W2[i];
  }
  if (t < USER_L) sbu[t] = bu[t];
  if (t < M1DIM)  sb1[t] = b1[t];
  if (t < M2DIM)  sb2[t] = b2[t];

  const int wave = t >> 5;
  const int lane = t & 31;
  const int ntiles = (B + 15) >> 4;
  int tile = blockIdx.x * WAVES + wave;
  if (tile >= ntiles) tile = ntiles - 1;        // duplicate work, same output
  const int r0 = tile * 16;
  float* buf = sbuf[wave];

  const int colh = lane & 15;   // A-frag row / B-frag column / D-frag column
  const int hi   = lane >> 4;   // 0 or 1 (lane half)
  const int koff = hi << 1;     // K sub-offset 0 or 2 (fp32 A/B frag layout)

  // ---- stage user rows (b128: each 16-lane half loads one full row) ----
  for (int i = 0; i < 16; i += 2) {
    int row = i + hi;
    int r = r0 + row; if (r >= B) r = B - 1;
    const int uid = user_idx[r];
    const v4f* src = (const v4f*)(user_emb + (size_t)uid * UID_DIM);
    *(v4f*)(buf + row * SSTR + 4 * colh) = src[colh];
  }
  __syncthreads();

  // ---------------- user GEMM: [16,64] x [64,32] -> uy ----------------
  v8f uy0 = {}; v8f uy1 = {};
  #pragma unroll
  for (int k = 0; k < UID_DIM; k += 4) {
    v2f a = *(const v2f*)(buf + colh * SSTR + k + koff);
    uy0 = wmma4(a, ldB(sWu, 64, k + koff, colh),      uy0);
    uy1 = wmma4(a, ldB(sWu, 64, k + koff, 16 + colh), uy1);
  }
  {
    float bc0 = sbu[colh], bc1 = sbu[16 + colh];
    #pragma unroll
    for (int r = 0; r < 8; ++r) {
      uy0[r] = fmaxf(uy0[r] + bc0, 0.f);
      uy1[r] = fmaxf(uy1[r] + bc1, 0.f);
    }
  }
  __syncthreads();

  // ---- stage movie_in: movie rows (cols 0..63) + cat sums (cols 64..79)
  for (int i = 0; i < 16; i += 2) {
    int row = i + hi;
    int r = r0 + row; if (r >= B) r = B - 1;
    const int mid = movie_idx[r];
    const v4f* src = (const v4f*)(movie_emb + (size_t)mid * MID_DIM);
    *(v4f*)(buf + row * SSTR + 4 * colh) = src[colh];
  }
  for (int i = 0; i < 16; i += 2) {
    int row = i + hi;
    int r = r0 + row; if (r >= B) r = B - 1;
    const int* cl = cat_lst + (size_t)r * MAX_CATS;
    float acc = 0.f;
    #pragma unroll
    for (int j = 0; j < MAX_CATS; ++j) {
      int ci = cl[j];
      if (ci >= 0) acc += cat_emb[ci * CAT_DIM + colh];
    }
    buf[row * SSTR + MID_DIM + colh] = acc;
  }
  __syncthreads();

  // ---------------- GEMM1: [16,80] x [80,64] -> h ----------------
  v8f h[4] = {};
  #pragma unroll
  for (int k = 0; k < MIN_DIM; k += 4) {
    v2f a = *(const v2f*)(buf + colh * SSTR + k + koff);
    #pragma unroll
    for (int n = 0; n < 4; ++n)
      h[n] = wmma4(a, ldB(sW1, 128, k + koff, n * 16 + colh), h[n]);
  }
  __syncthreads();
  // bias+relu, write h back to buf cols 0..63 row-major (D layout scatter)
  #pragma unroll
  for (int n = 0; n < 4; ++n) {
    float bc = sb1[n * 16 + colh];
    #pragma unroll
    for (int r = 0; r < 8; ++r) {
      float v = fmaxf(h[n][r] + bc, 0.f);
      buf[(r + hi * 8) * SSTR + n * 16 + colh] = v;
    }
  }
  __syncthreads();

  // ---------------- GEMM2: [16,64] x [64,32] -> my ----------------
  v8f my0 = {}; v8f my1 = {};
  #pragma unroll
  for (int k = 0; k < M1DIM; k += 4) {
    v2f a = *(const v2f*)(buf + colh * SSTR + k + koff);
    my0 = wmma4(a, ldB(sW2, 64, k + koff, colh),      my0);
    my1 = wmma4(a, ldB(sW2, 64, k + koff, 16 + colh), my1);
  }
  {
    float bc0 = sb2[colh], bc1 = sb2[16 + colh];
    #pragma unroll
    for (int r = 0; r < 8; ++r) {
      my0[r] = fmaxf(my0[r] + bc0, 0.f);
      my1[r] = fmaxf(my1[r] + bc1, 0.f);
    }
  }

  // ---------------- cosine similarity per row ----------------
  // D layout: VGPR r, lanes 0-15 = (row r, cols 0-15), lanes 16-31 = (row r+8)
  #pragma unroll
  for (int r = 0; r < 8; ++r) {
    float d  = uy0[r] * my0[r] + uy1[r] * my1[r];
    float un = uy0[r] * uy0[r] + uy1[r] * uy1[r];
    float mn = my0[r] * my0[r] + my1[r] * my1[r];
    #pragma unroll
    for (int m = 1; m < 16; m <<= 1) {   // reduce within each 16-lane half
      d  += __shfl_xor(d,  m, 32);
      un += __shfl_xor(un, m, 32);
      mn += __shfl_xor(mn, m, 32);
    }
    if (colh == 0) {
      int row = r0 + r + hi * 8;
      if (row < B) out[row] = d / (sqrtf(un) * sqrtf(mn));
    }
  }
}

extern "C" void kernel_launch(void* const* d_in, const int* in_sizes, int n_in,
                              void* d_out, int out_size, void* d_ws, size_t ws_size,
                              hipStream_t stream) {
  const float* user_emb  = (const float*)d_in[0];
  const float* movie_emb = (const float*)d_in[1];
  const float* cat_emb   = (const float*)d_in[2];
  const float* Wu        = (const float*)d_in[3];
  const float* bu        = (const float*)d_in[4];
  const float* W1        = (const float*)d_in[5];
  const float* b1        = (const float*)d_in[6];
  const float* W2        = (const float*)d_in[7];
  const float* b2        = (const float*)d_in[8];
  const int* user_idx    = (const int*)d_in[9];
  const int* movie_idx   = (const int*)d_in[10];
  const int* cat_lst     = (const int*)d_in[11];
  float* out = (float*)d_out;

  const int B = in_sizes[9];
  const int ntiles = (B + 15) / 16;
  const int blocks = (ntiles + WAVES - 1) / WAVES;
  twotower_kernel<<<blocks, TPB, 0, stream>>>(
      user_emb, movie_emb, cat_emb, Wu, bu, W1, b1, W2, b2,
      user_idx, movie_idx, cat_lst, out, B);
}